// BOWClassifier_20495583937151
// MI455X (gfx1250) — compile-verified
//
#include <hip/hip_runtime.h>
#include <hip/hip_bf16.h>

#define SEQ     512
#define BATCH   1024
#define EMB     300
#define HIDDEN  512
#define OUT_DIM 2

typedef __attribute__((ext_vector_type(2))) float v2f;
typedef __attribute__((ext_vector_type(8))) float v8f;

// ---------------------------------------------------------------------------
// Kernel 1: embedding gather + masked mean pooling.
// One block per batch element; thread e accumulates embedding dim e.
// Gathers mostly hit L2 (60MB table << 192MB L2).
// ---------------------------------------------------------------------------
__global__ __launch_bounds__(320) void bow_pool_kernel(
    const int* __restrict__ text,      // [SEQ, BATCH]
    const int* __restrict__ lengths,   // [BATCH]
    const float* __restrict__ emb,     // [VOCAB, EMB]
    float* __restrict__ pooled) {      // [BATCH, EMB]
  const int b = blockIdx.x;
  const int e = threadIdx.x;
  const int len = lengths[b];
  float acc = 0.0f;
  for (int s = 0; s < len; ++s) {
    // prefetch token stream ahead (emits global_prefetch_b8 on gfx1250)
    if (s + 64 < len) __builtin_prefetch(&text[(s + 64) * BATCH + b], 0, 0);
    const int tok = text[s * BATCH + b];      // uniform across block
    if (e < EMB) acc += emb[tok * EMB + e];   // coalesced gather row
  }
  if (e < EMB) pooled[b * EMB + e] = acc * (1.0f / (float)len);
}

// ---------------------------------------------------------------------------
// Kernel 2: h = relu(pooled @ W1 + b1) via V_WMMA_F32_16X16X4_F32.
// One wave32 per 16x16 output tile. K = 300 = 75 * 4 WMMA steps.
// A (16x4) lane layout: lane l -> row (l&15), K = 2*(l>>4) + {0,1}
// B (4x16) lane layout: lane l -> col (l&15), K = 2*(l>>4) + {0,1}
// C/D: VGPR j -> row j (lanes 0-15) / row j+8 (lanes 16-31), col = lane&15
// ---------------------------------------------------------------------------
__global__ __launch_bounds__(256) void bow_fc1_wmma_kernel(
    const float* __restrict__ pooled,  // [BATCH, EMB]
    const float* __restrict__ W1,      // [EMB, HIDDEN]
    const float* __restrict__ b1,      // [HIDDEN]
    float* __restrict__ h) {           // [BATCH, HIDDEN]
  const int wave = (blockIdx.x << 3) | (threadIdx.x >> 5);
  const int lane = threadIdx.x & 31;
  const int NT = HIDDEN / 16;              // 32 tiles along N
  const int tm = wave / NT;                // 0..63
  const int tn = wave % NT;                // 0..31
  const int m0 = tm * 16;
  const int n0 = tn * 16;
  const int li = lane & 15;
  const int half = lane >> 4;              // 0 or 1

  v8f c = {};
  // row base: (m0+li)*300 is a multiple of 4 elems, half*2 is even -> 8B aligned
  const float* __restrict__ arow = pooled + (m0 + li) * EMB + half * 2;
  const float* __restrict__ bcol = W1 + (half * 2) * HIDDEN + n0 + li;

#if __has_builtin(__builtin_amdgcn_wmma_f32_16x16x4_f32)
  for (int k = 0; k < EMB; k += 4) {
    v2f a = *(const v2f*)(arow + k);                 // A[m][k+2h], A[m][k+2h+1]
    v2f b;
    b.x = bcol[k * HIDDEN];                          // B[k+2h][n]
    b.y = bcol[k * HIDDEN + HIDDEN];                 // B[k+2h+1][n]
    // (neg_a, A, neg_b, B, c_mod, C, reuse_a, reuse_b)
    c = __builtin_amdgcn_wmma_f32_16x16x4_f32(false, a, false, b,
                                              (short)0, c, false, false);
  }
#else
  // Fallback (keeps file compiling if the builtin is absent; histogram will
  // show wmma=0 and we iterate): plain FMA accumulation in the same layout.
  for (int k = 0; k < EMB; k += 4) {
    v2f a = *(const v2f*)(arow + k);
    v2f b;
    b.x = bcol[k * HIDDEN];
    b.y = bcol[k * HIDDEN + HIDDEN];
    c[0] += a.x * b.x + a.y * b.y;  // placeholder, not layout-correct
  }
#endif

  const float bias = b1[n0 + li];
#pragma unroll
  for (int j = 0; j < 8; ++j) {
    float v = c[j] + bias;
    v = v > 0.0f ? v : 0.0f;                         // ReLU
    h[(m0 + j + half * 8) * HIDDEN + n0 + li] = v;
  }
}

// ---------------------------------------------------------------------------
// Kernel 3: out = h @ W2 + b2  (HIDDEN=512 -> OUT=2), one wave32 per row.
// ---------------------------------------------------------------------------
__global__ __launch_bounds__(256) void bow_fc2_kernel(
    const float* __restrict__ h,       // [BATCH, HIDDEN]
    const float* __restrict__ W2,      // [HIDDEN, OUT]
    const float* __restrict__ b2,      // [OUT]
    float* __restrict__ out) {         // [BATCH, OUT]
  const int row  = (blockIdx.x * blockDim.x + threadIdx.x) >> 5;
  const int lane = threadIdx.x & 31;
  if (row >= BATCH) return;
  const float* __restrict__ hr = h + row * HIDDEN;
  float a0 = 0.0f, a1 = 0.0f;
#pragma unroll 4
  for (int n = lane; n < HIDDEN; n += 32) {
    const float hv = hr[n];
    a0 += hv * W2[n * OUT_DIM + 0];
    a1 += hv * W2[n * OUT_DIM + 1];
  }
#pragma unroll
  for (int off = 16; off > 0; off >>= 1) {
    a0 += __shfl_xor(a0, off, 32);
    a1 += __shfl_xor(a1, off, 32);
  }
  if (lane == 0) {
    out[row * OUT_DIM + 0] = a0 + b2[0];
    out[row * OUT_DIM + 1] = a1 + b2[1];
  }
}

// ---------------------------------------------------------------------------
extern "C" void kernel_launch(void* const* d_in, const int* in_sizes, int n_in,
                              void* d_out, int out_size, void* d_ws, size_t ws_size,
                              hipStream_t stream) {
  const int*   text    = (const int*)d_in[0];     // [SEQ, BATCH]
  const int*   lengths = (const int*)d_in[1];     // [BATCH]
  const float* emb     = (const float*)d_in[2];   // [VOCAB, EMB]
  const float* W1      = (const float*)d_in[3];   // [EMB, HIDDEN]
  const float* b1      = (const float*)d_in[4];   // [HIDDEN]
  const float* W2      = (const float*)d_in[5];   // [HIDDEN, OUT]
  const float* b2      = (const float*)d_in[6];   // [OUT]
  float*       out     = (float*)d_out;           // [BATCH, OUT]

  float* pooled = (float*)d_ws;                   // BATCH*EMB floats
  float* hbuf   = pooled + (size_t)BATCH * EMB;   // BATCH*HIDDEN floats

  // Stage 1: gather + mean pool
  bow_pool_kernel<<<BATCH, 320, 0, stream>>>(text, lengths, emb, pooled);

  // Stage 2: FC1 + bias + ReLU via WMMA f32 16x16x4
  // tiles = (BATCH/16)*(HIDDEN/16) = 64*32 = 2048 waves; 8 waves/block
  bow_fc1_wmma_kernel<<<2048 / 8, 256, 0, stream>>>(pooled, W1, b1, hbuf);

  // Stage 3: FC2 + bias; 1024 waves, 8 waves/block
  bow_fc2_kernel<<<BATCH / 8, 256, 0, stream>>>(hbuf, W2, b2, out);
}